// BodyKinematics_40810779246606
// MI455X (gfx1250) — compile-verified
//
#include <hip/hip_runtime.h>
#include <hip/hip_bf16.h>

// CDNA5 / gfx1250. Batched tree-FK:
//   VALU/TRANS pipe : per-edge Rodrigues + 3x3 composition (massively parallel)
//   Matrix pipe     : tree combine via V_WMMA_F32_16X16X4_F32, four 4x4 affine
//                     products per instruction (block-diagonal extraction).
// One workgroup (8 wave32) per batch; locals/worlds staged in LDS with
// stride-17 rows (coprime with 64 banks -> conflict-free).

typedef float v2f __attribute__((ext_vector_type(2)));
typedef float v8f __attribute__((ext_vector_type(8)));

#define E_EDGES 255
#define N_NODES 256
#define LSTR 17              // floats per 4x4 in LDS (16 + 1 pad)
#define DUMMY_W (N_NODES * LSTR)                 // scratch slot (relative to wlds)
#define LDS_FLOATS (E_EDGES * LSTR + N_NODES * LSTR + 16)   // + dummy pad

__global__ __launch_bounds__(256, 4)
void fk_tree_wmma_kernel(const float* __restrict__ log_angles,     // (B, 3E)
                         const float* __restrict__ tip_to_base,    // (E,4,4)
                         const float* __restrict__ rot_axes,       // (3E,3)
                         const float* __restrict__ rot_constraints,// (3E,2)
                         float* __restrict__ out,                  // (B,N,4,4)
                         int B) {
    __shared__ float lds[LDS_FLOATS];
    float* llds = lds;                       // locals: edge e at e*LSTR, row-major 4x4
    float* wlds = lds + E_EDGES * LSTR;      // worlds: node i at i*LSTR, row-major 4x4

    const int b    = blockIdx.x;
    const int t    = threadIdx.x;
    const int lane = t & 31;
    const int wid  = t >> 5;
    const int half = lane >> 4;   // 0: K(0,1)/M(v) ; 1: K(2,3)/M(v+8)
    const int lr   = lane & 15;   // M row (A) == N col (B/D)

    // Pull this batch's angle slice toward the caches early (global_prefetch_b8).
    __builtin_prefetch(log_angles + (size_t)b * (3 * E_EDGES) + (size_t)t * 3, 0, 1);

    // ---------------- Phase A: per-edge local transforms (VALU) ----------------
    if (t < E_EDGES) {
        const int e = t;
        const float* la = log_angles + (size_t)b * (3 * E_EDGES) + 3 * e;

        float R[3][3][3];
#pragma unroll
        for (int i = 0; i < 3; ++i) {
            const int  a3 = 3 * e + i;
            const float sc  = rot_constraints[2 * a3 + 0];
            const float off = rot_constraints[2 * a3 + 1];
            float x = la[i];
#if __has_builtin(__builtin_amdgcn_tanhf)
            float th = __builtin_amdgcn_tanhf(x) * sc + off;
#else
            float th = tanhf(x) * sc + off;
#endif
            float s  = __sinf(th);
            float c1 = 1.0f - __cosf(th);
            float ax = rot_axes[3 * a3 + 0];
            float ay = rot_axes[3 * a3 + 1];
            float az = rot_axes[3 * a3 + 2];
            float n2 = ax * ax + ay * ay + az * az;   // K^2 = a a^T - |a|^2 I
            R[i][0][0] = 1.0f + c1 * (ax * ax - n2);
            R[i][0][1] = -s * az + c1 * (ax * ay);
            R[i][0][2] =  s * ay + c1 * (ax * az);
            R[i][1][0] =  s * az + c1 * (ay * ax);
            R[i][1][1] = 1.0f + c1 * (ay * ay - n2);
            R[i][1][2] = -s * ax + c1 * (ay * az);
            R[i][2][0] = -s * ay + c1 * (az * ax);
            R[i][2][1] =  s * ax + c1 * (az * ay);
            R[i][2][2] = 1.0f + c1 * (az * az - n2);
        }

        // r = R0 @ R1 @ R2
        float M1[3][3], r[3][3];
#pragma unroll
        for (int p = 0; p < 3; ++p)
#pragma unroll
            for (int q = 0; q < 3; ++q)
                M1[p][q] = R[1][p][0] * R[2][0][q] + R[1][p][1] * R[2][1][q] + R[1][p][2] * R[2][2][q];
#pragma unroll
        for (int p = 0; p < 3; ++p)
#pragma unroll
            for (int q = 0; q < 3; ++q)
                r[p][q] = R[0][p][0] * M1[0][q] + R[0][p][1] * M1[1][q] + R[0][p][2] * M1[2][q];

        // local = [r 0; 0 1] @ tip   (tip bottom row = [0,0,0,1])
        const float* tp = tip_to_base + (size_t)e * 16;
        float* Lp = llds + e * LSTR;
#pragma unroll
        for (int p = 0; p < 3; ++p)
#pragma unroll
            for (int q = 0; q < 4; ++q)
                Lp[p * 4 + q] = r[p][0] * tp[q] + r[p][1] * tp[4 + q] + r[p][2] * tp[8 + q];
        Lp[12] = 0.0f; Lp[13] = 0.0f; Lp[14] = 0.0f; Lp[15] = 1.0f;
    } else {
        // thread 255: root world = identity
#pragma unroll
        for (int j = 0; j < 16; ++j)
            wlds[j] = ((j & 3) == (j >> 2)) ? 1.0f : 0.0f;
    }

    // ---------------- Phase B: tree combine on the matrix pipe ----------------
    // Levels of the balanced tree (node order == BFS order), parent(i)=(i-1)>>1.
    const int lvl_start[8] = {1, 3, 7, 15, 31, 63, 127, 255};
    const int lvl_cnt[8]   = {2, 4, 8, 16, 32, 64, 128, 1};

    const int jblk = lr >> 2;          // 4x4 block index (A rows / B cols)
    const int rm   = lr & 3;           // row-within-block (A) / col (B,D)
    const int kb   = half << 1;        // K base: lanes 0-15 -> K0,1 ; 16-31 -> K2,3
    const bool act = (half == (jblk >> 1));  // lane holds diagonal-block rows
    const bool hi  = (jblk & 1) != 0;        // diag rows in d[4..7] vs d[0..3]

#pragma unroll 1
    for (int L = 0; L < 8; ++L) {
        __syncthreads();                       // prior level's worlds visible
        const int s0   = lvl_start[L];
        const int last = s0 + lvl_cnt[L] - 1;
        const int ngroups = (lvl_cnt[L] + 3) >> 2;
#pragma unroll 1
        for (int g = wid; g < ngroups; g += 8) {   // wave-uniform: EXEC all-1 at WMMA
            int id = s0 + 4 * g + jblk;
            if (id > last) id = last;              // clamp -> duplicate identical work
            const int par = (id - 1) >> 1;

            // A fragment (16x4): row M=lr of stacked parent worlds, cols K=kb,kb+1
            v2f afrag;
            afrag.x = wlds[par * LSTR + rm * 4 + kb];
            afrag.y = wlds[par * LSTR + rm * 4 + kb + 1];
            // B fragment (4x16): col N=lr of side-by-side locals, rows K=kb,kb+1
            v2f bfrag;
            bfrag.x = llds[(id - 1) * LSTR + kb * 4 + rm];
            bfrag.y = llds[(id - 1) * LSTR + (kb + 1) * 4 + rm];

            v8f acc = {};
#if __has_builtin(__builtin_amdgcn_wmma_f32_16x16x4_f32)
            v8f d = __builtin_amdgcn_wmma_f32_16x16x4_f32(
                        false, afrag, false, bfrag, (short)0, acc, false, false);
#else
#warning "wmma_f32_16x16x4_f32 builtin missing - VALU fallback in use"
            v8f d = acc;  // placeholder (compile-only fallback)
#endif
            // Diagonal block (jblk,jblk): constant-index extracts + cndmask selects.
            const float e0 = hi ? d[4] : d[0];
            const float e1 = hi ? d[5] : d[1];
            const float e2 = hi ? d[6] : d[2];
            const float e3 = hi ? d[7] : d[3];
            // Branchless store: inactive lanes dump into the LDS scratch slot.
            const int base = act ? (id * LSTR + rm) : DUMMY_W;
            wlds[base +  0] = e0;
            wlds[base +  4] = e1;
            wlds[base +  8] = e2;
            wlds[base + 12] = e3;
        }
    }
    __syncthreads();

    // ---------------- Phase C: coalesced output of all 256 worlds ----------------
    float* o = out + (size_t)b * (N_NODES * 16) + (size_t)t * 16;
    const float* w = wlds + t * LSTR;
#pragma unroll
    for (int q = 0; q < 4; ++q) {
        float4 v;
        v.x = w[q * 4 + 0];
        v.y = w[q * 4 + 1];
        v.z = w[q * 4 + 2];
        v.w = w[q * 4 + 3];
        reinterpret_cast<float4*>(o)[q] = v;
    }
}

extern "C" void kernel_launch(void* const* d_in, const int* in_sizes, int n_in,
                              void* d_out, int out_size, void* d_ws, size_t ws_size,
                              hipStream_t stream) {
    const float* log_angles      = (const float*)d_in[0];
    const float* tip_to_base     = (const float*)d_in[1];
    const float* rot_axes        = (const float*)d_in[2];
    const float* rot_constraints = (const float*)d_in[3];
    float* out = (float*)d_out;

    const int B = in_sizes[0] / (3 * E_EDGES);   // 4096
    fk_tree_wmma_kernel<<<dim3(B), dim3(256), 0, stream>>>(
        log_angles, tip_to_base, rot_axes, rot_constraints, out, B);
}